// Attenion_58248346468367
// MI455X (gfx1250) — compile-verified
//
#include <hip/hip_runtime.h>
#include <math.h>
#include <stdint.h>

// dims
#define NBATCH 32
#define NVARS  32
#define NHEAD  8
#define HTOK   64
#define DMODEL 128
#define HDIM   16
#define DFF    256
#define BN     (NBATCH * NVARS)       // 1024 slices
#define NROW   (BN * HTOK)            // 65536 token rows
#define NELEM  ((size_t)NROW * DMODEL)// 8,388,608 output elements
#define EPS    1e-5f

typedef float v2f __attribute__((ext_vector_type(2)));
typedef float v8f __attribute__((ext_vector_type(8)));

// ---------------- f32 WMMA 16x16x4 helpers ----------------
// A 16x4: lane<16 -> {A[r][0],A[r][1]}, lane>=16 -> {A[r][2],A[r][3]}
// B 4x16: lane<16 -> {B[0][n],B[1][n]}, lane>=16 -> {B[2][n],B[3][n]}
// C/D 16x16: VGPR r holds M = r + 8*(lane>=16), N = lane&15

__device__ __forceinline__ v8f wmma_step(v2f a, v2f b, v8f c) {
  return __builtin_amdgcn_wmma_f32_16x16x4_f32(false, a, false, b, (short)0, c,
                                               false, false);
}

// C += A(16xK) * B(Kx16),  A row-major lda, B row-major ldb
__device__ __forceinline__ v8f gemm_nn(const float* A, int lda, const float* B,
                                       int ldb, int K, v8f c) {
  int l = threadIdx.x & 31, hi = ((l >> 4) << 1), r = l & 15;
  for (int k = 0; k < K; k += 4) {
    v2f a, b;
    a.x = A[r * lda + k + hi];
    a.y = A[r * lda + k + hi + 1];
    b.x = B[(k + hi) * ldb + r];
    b.y = B[(k + hi + 1) * ldb + r];
    c = wmma_step(a, b, c);
  }
  return c;
}

// C += A(16xK) * Bt(16xK)^T   (Bt stored N-major: Bt[n][k])
__device__ __forceinline__ v8f gemm_nt(const float* A, int lda, const float* Bt,
                                       int ldbt, int K, v8f c) {
  int l = threadIdx.x & 31, hi = ((l >> 4) << 1), r = l & 15;
  for (int k = 0; k < K; k += 4) {
    v2f a, b;
    a.x = A[r * lda + k + hi];
    a.y = A[r * lda + k + hi + 1];
    b.x = Bt[r * ldbt + k + hi];
    b.y = Bt[r * ldbt + k + hi + 1];
    c = wmma_step(a, b, c);
  }
  return c;
}

// C += At(Kx16)^T * B(Kx16)   (At stored K-major: At[k][m])
__device__ __forceinline__ v8f gemm_tn(const float* At, int ldat, const float* B,
                                       int ldb, int K, v8f c) {
  int l = threadIdx.x & 31, hi = ((l >> 4) << 1), r = l & 15;
  for (int k = 0; k < K; k += 4) {
    v2f a, b;
    a.x = At[(k + hi) * ldat + r];
    a.y = At[(k + hi + 1) * ldat + r];
    b.x = B[(k + hi) * ldb + r];
    b.y = B[(k + hi + 1) * ldb + r];
    c = wmma_step(a, b, c);
  }
  return c;
}

__device__ __forceinline__ void store_tile(float* D, int ldd, v8f c) {
  int l = threadIdx.x & 31, hi = l >> 4, col = l & 15;
#pragma unroll
  for (int r = 0; r < 8; ++r) D[(r + 8 * hi) * ldd + col] = c[r];
}

// ---------------- gfx1250 async copy global->LDS (ASYNCcnt) ----------------
// GVS mode: mem_addr = SGPR base + 32-bit VGPR offset. VDST VGPR holds the
// LDS byte offset (low 32 bits of a generic pointer to __shared__).
__device__ __forceinline__ void async_ld_b128(uint32_t lds_off, uint32_t goff,
                                              unsigned long long gbase) {
  asm volatile("global_load_async_to_lds_b128 %0, %1, %2"
               :: "v"(lds_off), "v"(goff), "s"(gbase)
               : "memory");
}
__device__ __forceinline__ void wait_async0() {
  asm volatile("s_wait_asynccnt 0" ::: "memory");
}

// ---------------- kernel 0: zero stats ----------------
__global__ void k_zero(float* stats) {
  int i = threadIdx.x;
  if (i < 768) stats[i] = 0.0f;
  if (i + 256 < 768) stats[i + 256] = 0.0f;
  if (i + 512 < 768) stats[i + 512] = 0.0f;
}

// ---------------- kernel 1: qkv + dual attention per (slice, head) --------
__global__ __launch_bounds__(128) void k_attn(
    const float* __restrict__ src, const float* __restrict__ w_qkv,
    const float* __restrict__ b_qkv, float* __restrict__ m1,
    float* __restrict__ m2) {
  int slice = blockIdx.x >> 3;
  int head = blockIdx.x & 7;
  int tid = threadIdx.x;
  int wave = tid >> 5;
  int lane = tid & 31, hi = lane >> 4, col = lane & 15;

  __shared__ __align__(16) float s_src[HTOK * DMODEL]; // 32 KB (async staged)
  __shared__ __align__(16) float s_wq[DMODEL * 48];    // 24 KB (async staged)
  __shared__ float s_ema[HTOK * HTOK];                 // 16 KB
  __shared__ float s_q[HTOK * HDIM];                   // 4 KB
  __shared__ float s_k[HTOK * HDIM];
  __shared__ float s_v[HTOK * HDIM];
  __shared__ float s_eq[HTOK * HDIM];
  __shared__ float s_ek[HTOK * HDIM];
  __shared__ float s_st[HTOK * HTOK];                  // 16 KB (scores/probs)
  __shared__ float s_sh[HDIM * HDIM];                  // 1 KB

  // ---- issue async global->LDS copies, overlap with EMA construction ----
  {
    unsigned long long g_src =
        (unsigned long long)(uintptr_t)(src + (size_t)slice * HTOK * DMODEL);
    uint32_t l_src = (uint32_t)(uintptr_t)s_src;
    for (int i = tid; i < (HTOK * DMODEL) / 4; i += 128)  // 2048 x b128
      async_ld_b128(l_src + (uint32_t)i * 16u, (uint32_t)i * 16u, g_src);

    // stage the 48 qkv weight columns of this head: s_wq[k][sel*16+d]
    unsigned long long g_wq = (unsigned long long)(uintptr_t)w_qkv;
    uint32_t l_wq = (uint32_t)(uintptr_t)s_wq;
    for (int i = tid; i < DMODEL * 12; i += 128) {        // 1536 x b128
      int k = i / 12, rem = i % 12, sel = rem >> 2, c4 = rem & 3;
      uint32_t go = (uint32_t)((k * 384 + sel * DMODEL + head * HDIM + c4 * 4) * 4);
      uint32_t lo = (uint32_t)((k * 48 + sel * 16 + c4 * 4) * 4);
      async_ld_b128(l_wq + lo, go, g_wq);
    }
  }

  // EMA matrix, closed form: E[i][0]=(1-a)^i, E[i][j]=a*(1-a)^(i-j), j<=i
  for (int idx = tid; idx < HTOK * HTOK; idx += 128) {
    int i = idx >> 6, j = idx & 63;
    float v = 0.0f;
    if (j <= i)
      v = (j == 0) ? __powf(0.1f, (float)i) : 0.9f * __powf(0.1f, (float)(i - j));
    s_ema[idx] = v;
  }
  wait_async0();
  __syncthreads();

  // qkv sub-GEMM from LDS: (64x128)@(128x48); wave w does row-tile w
  {
    int mt = wave;
    for (int sel = 0; sel < 3; ++sel) {
      v8f c = {};
      c = gemm_nn(s_src + mt * 16 * DMODEL, DMODEL, s_wq + sel * 16, 48, DMODEL, c);
      float bias = b_qkv[sel * DMODEL + head * HDIM + col];
      float* dst = (sel == 0) ? s_q : (sel == 1) ? s_k : s_v;
#pragma unroll
      for (int r = 0; r < 8; ++r)
        dst[(mt * 16 + r + 8 * hi) * HDIM + col] = c[r] + bias;
    }
  }
  __syncthreads();

  // eq = EMA @ q, ek = EMA @ k  (wave w does row-tile w of each)
  {
    int mt = wave;
    v8f c = {};
    c = gemm_nn(s_ema + mt * 16 * HTOK, HTOK, s_q, HDIM, HTOK, c);
    store_tile(s_eq + mt * 16 * HDIM, HDIM, c);
    v8f d = {};
    d = gemm_nn(s_ema + mt * 16 * HTOK, HTOK, s_k, HDIM, HTOK, d);
    store_tile(s_ek + mt * 16 * HDIM, HDIM, d);
  }
  __syncthreads();

  // s_tok = eq @ ek^T * sqrt(HD); wave w does column-tile w for all row-tiles
  for (int mt = 0; mt < 4; ++mt) {
    int nt = wave;
    v8f c = {};
    c = gemm_nt(s_eq + mt * 16 * HDIM, HDIM, s_ek + nt * 16 * HDIM, HDIM, HDIM, c);
#pragma unroll
    for (int r = 0; r < 8; ++r)
      s_st[(mt * 16 + r + 8 * hi) * HTOK + nt * 16 + col] = c[r] * 4.0f;
  }
  __syncthreads();

  // softmax over rows of s_st (64 rows; threads 0..63 = whole waves 0-1)
  if (tid < HTOK) {
    float* row = s_st + tid * HTOK;
    float m = row[0];
    for (int j = 1; j < HTOK; ++j) m = fmaxf(m, row[j]);
    float s = 0.0f;
    for (int j = 0; j < HTOK; ++j) {
      float e = __expf(row[j] - m);
      row[j] = e;
      s += e;
    }
    float inv = 1.0f / s;
    for (int j = 0; j < HTOK; ++j) row[j] *= inv;
  }
  __syncthreads();

  // o_tok = a_tok @ v, scatter through merge permutation
  float* m1s = m1 + (size_t)slice * HTOK * DMODEL;
  {
    int mt = wave;
    v8f c = {};
    c = gemm_nn(s_st + mt * 16 * HTOK, HTOK, s_v, HDIM, HTOK, c);
#pragma unroll
    for (int r = 0; r < 8; ++r) {
      int tok = mt * 16 + r + 8 * hi;
      int tau = (head & 1) * 32 + (tok >> 1);
      int ch = (tok & 1) * 64 + (head >> 1) * 16 + col;
      m1s[tau * DMODEL + ch] = c[r];
    }
  }

  // s_hid = q^T @ k * sqrt(H)  (one 16x16 tile; all waves compute identical
  // values and store them -> WMMA always under full EXEC)
  {
    v8f c = {};
    c = gemm_tn(s_q, HDIM, s_k, HDIM, HTOK, c);
#pragma unroll
    for (int r = 0; r < 8; ++r)
      s_sh[(r + 8 * hi) * HDIM + col] = c[r] * 8.0f;
  }
  __syncthreads();

  // softmax over 16 rows of s_sh
  if (tid < HDIM) {
    float* row = s_sh + tid * HDIM;
    float m = row[0];
    for (int j = 1; j < HDIM; ++j) m = fmaxf(m, row[j]);
    float s = 0.0f;
    for (int j = 0; j < HDIM; ++j) {
      float e = __expf(row[j] - m);
      row[j] = e;
      s += e;
    }
    float inv = 1.0f / s;
    for (int j = 0; j < HDIM; ++j) row[j] *= inv;
  }
  __syncthreads();

  // o_hid = v @ a_hid^T, scatter through merge permutation
  float* m2s = m2 + (size_t)slice * HTOK * DMODEL;
  {
    int mt = wave;
    v8f c = {};
    c = gemm_nt(s_v + mt * 16 * HDIM, HDIM, s_sh, HDIM, HDIM, c);
#pragma unroll
    for (int r = 0; r < 8; ++r) {
      int tok = mt * 16 + r + 8 * hi;
      int tau = (head & 1) * 32 + (tok >> 1);
      int ch = (tok & 1) * 64 + (head >> 1) * 16 + col;
      m2s[tau * DMODEL + ch] = c[r];
    }
  }
}

// ---------------- kernel 2: BN batch stats for m1, m2 ----------------
__global__ __launch_bounds__(128) void k_stats(const float* __restrict__ m1,
                                               const float* __restrict__ m2,
                                               float* __restrict__ stats) {
  int ch = threadIdx.x;                 // 0..127
  size_t r0 = (size_t)blockIdx.x * 256; // 256 rows per block
  float s1 = 0, q1 = 0, s2 = 0, q2 = 0;
  for (int r = 0; r < 256; ++r) {
    float a = m1[(r0 + r) * DMODEL + ch];
    float b = m2[(r0 + r) * DMODEL + ch];
    s1 += a; q1 += a * a;
    s2 += b; q2 += b * b;
  }
  atomicAdd(&stats[ch], s1);
  atomicAdd(&stats[DMODEL + ch], q1);
  atomicAdd(&stats[2 * DMODEL + ch], s2);
  atomicAdd(&stats[3 * DMODEL + ch], q2);
}

// ---------------- kernel 3: BN apply + dual FFN + residual + stats3 -------
__global__ __launch_bounds__(128) void k_ffn(
    const float* __restrict__ src, const float* __restrict__ m1,
    const float* __restrict__ m2, const float* __restrict__ stats,
    const float* __restrict__ w1a, const float* __restrict__ b1a,
    const float* __restrict__ w2a, const float* __restrict__ b2a,
    const float* __restrict__ w1b, const float* __restrict__ b1b,
    const float* __restrict__ w2b, const float* __restrict__ b2b,
    const float* __restrict__ g1, const float* __restrict__ be1,
    const float* __restrict__ g2, const float* __restrict__ be2,
    float* __restrict__ out, float* __restrict__ stats_out) {
  int tid = threadIdx.x;
  int wave = tid >> 5;
  int lane = tid & 31, hi = lane >> 4, col = lane & 15;
  size_t r0 = (size_t)blockIdx.x * 64;

  __shared__ float s_x1[64 * DMODEL];  // 32 KB (norm(m1), later ff1)
  __shared__ float s_x2[64 * DMODEL];  // 32 KB (norm(m2))
  __shared__ float s_h[64 * DFF];      // 64 KB (gelu hidden)
  __shared__ float s_sc1[DMODEL], s_sh1[DMODEL];
  __shared__ float s_sc2[DMODEL], s_sh2[DMODEL];
  __shared__ float s_sum[DMODEL], s_ssq[DMODEL];

  // warm GL2 for the weight matrices (global_prefetch_b8, one line per lane)
  __builtin_prefetch(w1a + (size_t)tid * 128, 0, 1);
  __builtin_prefetch(w1b + (size_t)tid * 128, 0, 1);
  __builtin_prefetch(w2a + (size_t)tid * 128, 0, 1);
  __builtin_prefetch(w2b + (size_t)tid * 128, 0, 1);

  // per-channel scale/shift from batch stats; zero local stats3 partials
  {
    const float invn = 1.0f / (float)NROW;
    float mu1 = stats[tid] * invn;
    float var1 = stats[DMODEL + tid] * invn - mu1 * mu1;
    float sc1 = g1[tid] * rsqrtf(var1 + EPS);
    s_sc1[tid] = sc1;
    s_sh1[tid] = be1[tid] - mu1 * sc1;
    float mu2 = stats[2 * DMODEL + tid] * invn;
    float var2 = stats[3 * DMODEL + tid] * invn - mu2 * mu2;
    float sc2 = g2[tid] * rsqrtf(var2 + EPS);
    s_sc2[tid] = sc2;
    s_sh2[tid] = be2[tid] - mu2 * sc2;
    s_sum[tid] = 0.0f;
    s_ssq[tid] = 0.0f;
  }
  __syncthreads();

  // normalize m1 -> s_x1, m2 -> s_x2
  for (int idx = tid; idx < 64 * DMODEL; idx += 128) {
    int r = idx >> 7, ch = idx & 127;
    s_x1[idx] = m1[(r0 + r) * DMODEL + ch] * s_sc1[ch] + s_sh1[ch];
    s_x2[idx] = m2[(r0 + r) * DMODEL + ch] * s_sc2[ch] + s_sh2[ch];
  }
  __syncthreads();

  // ---- branch a: h = gelu(x1 @ w1a + b1a) ----
  for (int t = wave; t < 64; t += 4) {  // 4 mt x 16 nt
    int mt = t >> 4, nt = t & 15;
    v8f c = {};
    c = gemm_nn(s_x1 + mt * 16 * DMODEL, DMODEL, w1a + nt * 16, DFF, DMODEL, c);
    float bias = b1a[nt * 16 + col];
#pragma unroll
    for (int r = 0; r < 8; ++r) {
      float x = c[r] + bias;
      s_h[(mt * 16 + r + 8 * hi) * DFF + nt * 16 + col] =
          0.5f * x * (1.0f + erff(x * 0.70710678118654752f));
    }
  }
  __syncthreads();
  // ff1 = h @ w2a + b2a  -> overwrite s_x1
  for (int t = wave; t < 32; t += 4) {  // 4 mt x 8 nt
    int mt = t >> 3, nt = t & 7;
    v8f c = {};
    c = gemm_nn(s_h + mt * 16 * DFF, DFF, w2a + nt * 16, DMODEL, DFF, c);
    float bias = b2a[nt * 16 + col];
#pragma unroll
    for (int r = 0; r < 8; ++r)
      s_x1[(mt * 16 + r + 8 * hi) * DMODEL + nt * 16 + col] = c[r] + bias;
  }
  __syncthreads();

  // ---- branch b: h = gelu(x2 @ w1b + b1b) ----
  for (int t = wave; t < 64; t += 4) {
    int mt = t >> 4, nt = t & 15;
    v8f c = {};
    c = gemm_nn(s_x2 + mt * 16 * DMODEL, DMODEL, w1b + nt * 16, DFF, DMODEL, c);
    float bias = b1b[nt * 16 + col];
#pragma unroll
    for (int r = 0; r < 8; ++r) {
      float x = c[r] + bias;
      s_h[(mt * 16 + r + 8 * hi) * DFF + nt * 16 + col] =
          0.5f * x * (1.0f + erff(x * 0.70710678118654752f));
    }
  }
  __syncthreads();
  // x = src + ff1 + (h @ w2b + b2b); write pre-BN x, accumulate stats3
  for (int t = wave; t < 32; t += 4) {
    int mt = t >> 3, nt = t & 7;
    v8f c = {};
    c = gemm_nn(s_h + mt * 16 * DFF, DFF, w2b + nt * 16, DMODEL, DFF, c);
    float bias = b2b[nt * 16 + col];
#pragma unroll
    for (int r = 0; r < 8; ++r) {
      int row = mt * 16 + r + 8 * hi;
      int ch = nt * 16 + col;
      float x = c[r] + bias + s_x1[row * DMODEL + ch] +
                src[(r0 + row) * DMODEL + ch];
      out[(r0 + row) * DMODEL + ch] = x;
      atomicAdd(&s_sum[ch], x);
      atomicAdd(&s_ssq[ch], x * x);
    }
  }
  __syncthreads();
  atomicAdd(&stats_out[4 * DMODEL + tid], s_sum[tid]);
  atomicAdd(&stats_out[5 * DMODEL + tid], s_ssq[tid]);
}

// ---------------- kernel 4: final BN, in place on d_out ----------------
__global__ __launch_bounds__(256) void k_bnout(float* __restrict__ out,
                                               const float* __restrict__ stats,
                                               const float* __restrict__ g3,
                                               const float* __restrict__ be3) {
  size_t i = (size_t)blockIdx.x * 256 + threadIdx.x;
  int ch = (int)(i & 127);
  const float invn = 1.0f / (float)NROW;
  float mu = stats[4 * DMODEL + ch] * invn;
  float var = stats[5 * DMODEL + ch] * invn - mu * mu;
  out[i] = (out[i] - mu) * rsqrtf(var + EPS) * g3[ch] + be3[ch];
}

extern "C" void kernel_launch(void* const* d_in, const int* in_sizes, int n_in,
                              void* d_out, int out_size, void* d_ws,
                              size_t ws_size, hipStream_t stream) {
  (void)in_sizes; (void)n_in; (void)out_size; (void)ws_size;
  const float* src   = (const float*)d_in[0];
  const float* w_qkv = (const float*)d_in[1];
  const float* b_qkv = (const float*)d_in[2];
  const float* w1a   = (const float*)d_in[3];
  const float* b1a   = (const float*)d_in[4];
  const float* w2a   = (const float*)d_in[5];
  const float* b2a   = (const float*)d_in[6];
  const float* w1b   = (const float*)d_in[7];
  const float* b1b   = (const float*)d_in[8];
  const float* w2b   = (const float*)d_in[9];
  const float* b2b   = (const float*)d_in[10];
  const float* g1    = (const float*)d_in[11];
  const float* be1   = (const float*)d_in[12];
  const float* g2    = (const float*)d_in[13];
  const float* be2   = (const float*)d_in[14];
  const float* g3    = (const float*)d_in[15];
  const float* be3   = (const float*)d_in[16];
  float* out = (float*)d_out;

  float* m1 = (float*)d_ws;            // 8,388,608 floats
  float* m2 = m1 + NELEM;              // 8,388,608 floats
  float* stats = m2 + NELEM;           // 768 floats

  k_zero<<<1, 256, 0, stream>>>(stats);
  k_attn<<<BN * NHEAD, 128, 0, stream>>>(src, w_qkv, b_qkv, m1, m2);
  k_stats<<<NROW / 256, 128, 0, stream>>>(m1, m2, stats);
  k_ffn<<<NROW / 64, 128, 0, stream>>>(src, m1, m2, stats, w1a, b1a, w2a, b2a,
                                       w1b, b1b, w2b, b2b, g1, be1, g2, be2,
                                       out, stats);
  k_bnout<<<(unsigned)(NELEM / 256), 256, 0, stream>>>(out, stats, g3, be3);
}